// transformer_HF_30983894073437
// MI455X (gfx1250) — compile-verified
//
#include <hip/hip_runtime.h>
#include <math.h>

// ---------------------------------------------------------------- constants
static constexpr int CB    = 32;     // batch
static constexpr int CS    = 512;    // seq
static constexpr int CIN   = 512;
static constexpr int CH    = 512;
static constexpr int CNH   = 4;
static constexpr int CHD   = 128;
static constexpr int CINT  = 256;
static constexpr int CGH   = 256;    // H/2
static constexpr int CSPAN = 256;
static constexpr int CBS   = CB * CS;              // 16384 rows
#define INV_SCALE 0.051031036307982884f            // 1/sqrt(128*3)

typedef __attribute__((ext_vector_type(2))) float v2f;
typedef __attribute__((ext_vector_type(8))) float v8f;

// ---------------------------------------------------------------- WMMA strip
// One wave computes a 16 x (16*NT) f32 strip via V_WMMA_F32_16X16X4_F32,
// reusing the A fragment across NT B tiles (4 WMMAs per A load for NT=4).
// A tile pointer already offset to (m,0); B pointer offset to (0,n) or (n,0)
// if BT (B given row-major as [N][K], i.e. we compute A @ B^T).
template<bool BT, int NT>
__device__ __forceinline__ void wmma_strip(const float* __restrict__ A, int lda,
                                           const float* __restrict__ Bm, int ldb,
                                           int K, int lh, int ll, v8f* acc)
{
    const float* arow = A + (size_t)ll * lda + 2 * lh;        // A[m+ll][2*lh + k]
    const float* brow[NT];
#pragma unroll
    for (int t = 0; t < NT; ++t) {
        if (BT) brow[t] = Bm + (size_t)(t * 16 + ll) * ldb + 2 * lh; // B[n+t*16+ll][2lh+k]
        else    brow[t] = Bm + (size_t)(2 * lh) * ldb + t * 16 + ll; // B[2lh+k][n+t*16+ll]
    }
#pragma unroll 2
    for (int k = 0; k < K; k += 4) {
        v2f a = *(const v2f*)(arow + k);
#pragma unroll
        for (int t = 0; t < NT; ++t) {
            v2f b;
            if (BT) {
                b = *(const v2f*)(brow[t] + k);
            } else {
                b.x = brow[t][(size_t)k * ldb];
                b.y = brow[t][(size_t)(k + 1) * ldb];
            }
            acc[t] = __builtin_amdgcn_wmma_f32_16x16x4_f32(false, a, false, b,
                                                           (short)0, acc[t], false, false);
        }
    }
}

__device__ __forceinline__ float gelu_exact(float x)
{
    return 0.5f * x * (1.0f + erff(x * 0.70710678118654752f));
}

// ---------------------------------------------------------------- generic GEMM
// C[M,N] = A[M,K] @ op(B) (+bias[N]) (+resid, same layout as C) (+GELU)
// grid: (M/16, N/(64*NT)), block 128 (4 waves, each one 16x(16*NT) strip)
template<bool BT, bool BIAS, bool RESID, int ACT, int NT>
__global__ __launch_bounds__(128)
void k_gemm(const float* __restrict__ A, const float* __restrict__ Bm,
            const float* __restrict__ bias, const float* __restrict__ resid,
            float* __restrict__ C, int K, int lda, int ldb, int ldc)
{
    const int l  = threadIdx.x & 31;
    const int w  = threadIdx.x >> 5;
    const int lh = l >> 4, ll = l & 15;
    const int m  = blockIdx.x * 16;
    const int n  = (blockIdx.y * 4 + w) * (16 * NT);

    const float* At = A + (size_t)m * lda;
    const float* Bt = BT ? (Bm + (size_t)n * ldb) : (Bm + n);
    v8f acc[NT];
#pragma unroll
    for (int t = 0; t < NT; ++t) acc[t] = (v8f){};
    wmma_strip<BT, NT>(At, lda, Bt, ldb, K, lh, ll, acc);

#pragma unroll
    for (int t = 0; t < NT; ++t) {
        const int nn = n + t * 16 + ll;
        const float bv = BIAS ? bias[nn] : 0.0f;
#pragma unroll
        for (int i = 0; i < 8; ++i) {
            const int mm = m + i + 8 * lh;
            float vv = acc[t][i] + bv;
            if (RESID) vv += resid[(size_t)mm * ldc + nn];
            if (ACT == 1) vv = gelu_exact(vv);
            C[(size_t)mm * ldc + nn] = vv;
        }
    }
}

// ---------------------------------------------------------------- GRU scan
// One workgroup (256 thr) per (batch, direction); hidden state in LDS.
__global__ __launch_bounds__(256)
void k_gru_scan(const float* __restrict__ xpF, const float* __restrict__ xpB,
                const float* __restrict__ WhhF, const float* __restrict__ bhhF,
                const float* __restrict__ WhhB, const float* __restrict__ bhhB,
                float* __restrict__ out)
{
    const int b   = blockIdx.x;
    const int dir = blockIdx.y;
    const float* xp  = dir ? xpB  : xpF;
    const float* Whh = dir ? WhhB : WhhF;
    const float* bhh = dir ? bhhB : bhhF;

    __shared__ float hs[CGH];
    const int j = threadIdx.x;
    hs[j] = 0.0f;
    __syncthreads();

    const float* wr = Whh + (size_t)j * CGH;
    const float* wz = Whh + (size_t)(j + CGH) * CGH;
    const float* wn = Whh + (size_t)(j + 2 * CGH) * CGH;
    const float br = bhh[j], bz = bhh[j + CGH], bn = bhh[j + 2 * CGH];

    for (int t = 0; t < CS; ++t) {
        const int tt = dir ? (CS - 1 - t) : t;
        const float* xg = xp + ((size_t)b * CS + tt) * (3 * CGH);
        // prefetch next timestep's gate inputs (global_prefetch_b8)
        if (t + 1 < CS) {
            const int tn = dir ? (CS - 2 - t) : (t + 1);
            __builtin_prefetch(xp + ((size_t)b * CS + tn) * (3 * CGH) + j, 0, 1);
        }
        float hr = br, hz = bz, hn = bn;
#pragma unroll 8
        for (int kk = 0; kk < CGH; ++kk) {
            const float hv = hs[kk];
            hr = fmaf(wr[kk], hv, hr);
            hz = fmaf(wz[kk], hv, hz);
            hn = fmaf(wn[kk], hv, hn);
        }
        const float hp = hs[j];
        const float r = 1.0f / (1.0f + expf(-(xg[j] + hr)));
        const float z = 1.0f / (1.0f + expf(-(xg[j + CGH] + hz)));
        const float n = tanhf(xg[j + 2 * CGH] + r * hn);
        const float hnew = (1.0f - z) * n + z * hp;
        __syncthreads();
        hs[j] = hnew;
        out[((size_t)b * CS + tt) * CH + dir * CGH + j] = hnew;
        __syncthreads();
    }
}

// ---------------------------------------------------------------- layernorm (width 512)
__global__ __launch_bounds__(256)
void k_layernorm(const float* __restrict__ in, float* __restrict__ out,
                 const float* __restrict__ g, const float* __restrict__ bb)
{
    __shared__ float s1[256], s2[256];
    const int t = threadIdx.x;
    const float* r = in + (size_t)blockIdx.x * CH;
    const float a0 = r[t], a1 = r[t + 256];
    s1[t] = a0 + a1;
    s2[t] = a0 * a0 + a1 * a1;
    __syncthreads();
    for (int o = 128; o > 0; o >>= 1) {
        if (t < o) { s1[t] += s1[t + o]; s2[t] += s2[t + o]; }
        __syncthreads();
    }
    const float mu  = s1[0] * (1.0f / 512.0f);
    const float var = s2[0] * (1.0f / 512.0f) - mu * mu;
    const float rs  = rsqrtf(var + 1e-7f);
    float* o = out + (size_t)blockIdx.x * CH;
    o[t]       = (a0 - mu) * rs * g[t]       + bb[t];
    o[t + 256] = (a1 - mu) * rs * g[t + 256] + bb[t + 256];
}

// ---------------------------------------------------------------- softmax (width 512)
__global__ __launch_bounds__(512)
void k_softmax(float* __restrict__ sc)
{
    __shared__ float red[512];
    const int t = threadIdx.x;
    float* r = sc + (size_t)blockIdx.x * CS;
    const float v = r[t];
    red[t] = v; __syncthreads();
    for (int o = 256; o > 0; o >>= 1) {
        if (t < o) red[t] = fmaxf(red[t], red[t + o]);
        __syncthreads();
    }
    const float mx = red[0];
    __syncthreads();
    const float e = expf(v - mx);
    red[t] = e; __syncthreads();
    for (int o = 256; o > 0; o >>= 1) {
        if (t < o) red[t] += red[t + o];
        __syncthreads();
    }
    r[t] = e / red[0];
}

// ---------------------------------------------------------------- rel-pos buckets
__global__ __launch_bounds__(256)
void k_relpos(int* __restrict__ rp)
{
    const int idx = blockIdx.x * 256 + threadIdx.x;
    const int i = idx >> 9, jc = idx & 511;
    const int rel = i - jc;
    const int mid = CSPAN / 2;                       // 128
    const float sgn = (rel > 0) ? 1.0f : ((rel < 0) ? -1.0f : 0.0f);
    const int a = (rel < mid && rel > -mid) ? (mid - 1) : abs(rel);
    const float logp = ceilf(logf((float)a / (float)mid) /
                             logf(511.0f / 128.0f) * (float)(mid - 1)) + (float)mid;
    const float bucket = (a <= mid) ? (float)rel : logp * sgn;
    rp[idx] = (int)bucket;
}

// ---------------------------------------------------------------- c2p / p2c
// OUT[z][s][p] = QK[b, s, h,:] . POS[p, h,:]   z = b*NH + h
// grid (32, 2, 128), block 128; each wave a 16x64 strip (NT=4)
__global__ __launch_bounds__(128)
void k_qpos(const float* __restrict__ QK, const float* __restrict__ POS,
            float* __restrict__ OUT)
{
    const int z = blockIdx.z, b = z >> 2, h = z & 3;
    const int l = threadIdx.x & 31, w = threadIdx.x >> 5;
    const int lh = l >> 4, ll = l & 15;
    const int m = blockIdx.x * 16;
    const int n = (blockIdx.y * 4 + w) * 64;
    const float* A  = QK  + ((size_t)b * CS + m) * CH + h * CHD;
    const float* Bt = POS + (size_t)n * CH + h * CHD;
    v8f acc[4];
#pragma unroll
    for (int t = 0; t < 4; ++t) acc[t] = (v8f){};
    wmma_strip<true, 4>(A, CH, Bt, CH, CHD, lh, ll, acc);
#pragma unroll
    for (int t = 0; t < 4; ++t) {
        const int nn = n + t * 16 + ll;
#pragma unroll
        for (int i = 0; i < 8; ++i) {
            const int mm = m + i + 8 * lh;
            OUT[((size_t)z * CS + mm) * 512 + nn] = acc[t][i];
        }
    }
}

// ---------------------------------------------------------------- scores
// sc[z][q][k] = ( q.k + c2p[z,q, clamp(rp[q,k]+256)] + p2c[z,k, clamp(256-rp[k,q])] ) / SCALE
__global__ __launch_bounds__(128)
void k_scores(const float* __restrict__ Q, const float* __restrict__ Kt,
              const int* __restrict__ rp, const float* __restrict__ c2p,
              const float* __restrict__ p2c, float* __restrict__ sc)
{
    const int z = blockIdx.z, b = z >> 2, h = z & 3;
    const int l = threadIdx.x & 31, w = threadIdx.x >> 5;
    const int lh = l >> 4, ll = l & 15;
    const int m = blockIdx.x * 16;
    const int n = (blockIdx.y * 4 + w) * 64;
    const float* A  = Q  + ((size_t)b * CS + m) * CH + h * CHD;
    const float* Bt = Kt + ((size_t)b * CS + n) * CH + h * CHD;
    v8f acc[4];
#pragma unroll
    for (int t = 0; t < 4; ++t) acc[t] = (v8f){};
    wmma_strip<true, 4>(A, CH, Bt, CH, CHD, lh, ll, acc);
#pragma unroll
    for (int t = 0; t < 4; ++t) {
        const int nn = n + t * 16 + ll;
#pragma unroll
        for (int i = 0; i < 8; ++i) {
            const int mm = m + i + 8 * lh;
            const int r1 = rp[mm * 512 + nn];
            const int r2 = rp[nn * 512 + mm];
            const int ci = min(max(r1 + 256, 0), 511);
            const int pi = min(max(256 - r2, 0), 511);
            const float g1 = c2p[((size_t)z * CS + mm) * 512 + ci];
            const float g2 = p2c[((size_t)z * CS + nn) * 512 + pi];
            sc[((size_t)z * CS + mm) * CS + nn] = (acc[t][i] + g1 + g2) * INV_SCALE;
        }
    }
}

// ---------------------------------------------------------------- ctx = probs @ v
// grid (32, 1, 128), block 128; each wave a 16x32 strip (NT=2);
// writes directly in (b,s,H) layout
__global__ __launch_bounds__(128)
void k_ctx(const float* __restrict__ sc, const float* __restrict__ V,
           float* __restrict__ ctx)
{
    const int z = blockIdx.z, b = z >> 2, h = z & 3;
    const int l = threadIdx.x & 31, w = threadIdx.x >> 5;
    const int lh = l >> 4, ll = l & 15;
    const int m = blockIdx.x * 16;
    const int n = w * 32;
    const float* A  = sc + ((size_t)z * CS + m) * CS;
    const float* Bt = V + (size_t)b * CS * CH + h * CHD + n;      // [k][n], ldb=CH
    v8f acc[2];
    acc[0] = (v8f){}; acc[1] = (v8f){};
    wmma_strip<false, 2>(A, CS, Bt, CH, CS, lh, ll, acc);
#pragma unroll
    for (int t = 0; t < 2; ++t) {
        const int nn = n + t * 16 + ll;
#pragma unroll
        for (int i = 0; i < 8; ++i) {
            const int mm = m + i + 8 * lh;
            ctx[((size_t)b * CS + mm) * CH + h * CHD + nn] = acc[t][i];
        }
    }
}

// ================================================================ launch
extern "C" void kernel_launch(void* const* d_in, const int* in_sizes, int n_in,
                              void* d_out, int out_size, void* d_ws, size_t ws_size,
                              hipStream_t stream)
{
    const float* X = (const float*)d_in[0];
    const float *gWih[4], *gWhh[4], *gbih[4], *gbhh[4];
    for (int i = 0; i < 4; ++i) {
        gWih[i] = (const float*)d_in[1 + 4 * i];
        gWhh[i] = (const float*)d_in[2 + 4 * i];
        gbih[i] = (const float*)d_in[3 + 4 * i];
        gbhh[i] = (const float*)d_in[4 + 4 * i];
    }
    const float* rel_emb = (const float*)d_in[17];
    const float* rel_g   = (const float*)d_in[18];
    const float* rel_b   = (const float*)d_in[19];

    float* ws = (float*)d_ws;
    size_t off = 0;
    auto alloc = [&](size_t n) -> float* { float* p = ws + off; off += (n + 15) & ~(size_t)15; return p; };

    float* xpF    = alloc((size_t)CBS * 768);
    float* xpB    = alloc((size_t)CBS * 768);
    float* gx1    = alloc((size_t)CBS * CH);
    float* gx2    = alloc((size_t)CBS * CH);
    float* relln  = alloc((size_t)512 * CH);
    int*   relpos = (int*)alloc((size_t)512 * 512);
    float* q      = alloc((size_t)CBS * CH);
    float* kk     = alloc((size_t)CBS * CH);
    float* v      = alloc((size_t)CBS * CH);
    float* posk   = alloc((size_t)512 * CH);
    float* posq   = alloc((size_t)512 * CH);
    float* c2p    = alloc((size_t)CB * CNH * CS * 512);
    float* p2c    = alloc((size_t)CB * CNH * CS * 512);
    float* scores = alloc((size_t)CB * CNH * CS * CS);
    float* ctx    = alloc((size_t)CBS * CH);
    float* y1     = alloc((size_t)CBS * CH);
    float* a1     = alloc((size_t)CBS * CH);
    float* ff     = alloc((size_t)CBS * CINT);
    float* y2     = alloc((size_t)CBS * CH);
    float* hbuf   = alloc((size_t)CBS * CH);

    const dim3 blk128(128), blk256(256), blk512(512);

    // ---- GRU layer 0 (input X, K=512) : xp = x @ Wih^T + bih   (N=768 -> grid.y=3)
    k_gemm<true, true, false, 0, 4><<<dim3(CBS / 16, 3), blk128, 0, stream>>>(
        X, gWih[0], gbih[0], nullptr, xpF, CIN, CIN, CIN, 768);
    k_gemm<true, true, false, 0, 4><<<dim3(CBS / 16, 3), blk128, 0, stream>>>(
        X, gWih[1], gbih[1], nullptr, xpB, CIN, CIN, CIN, 768);
    k_gru_scan<<<dim3(CB, 2), blk256, 0, stream>>>(xpF, xpB, gWhh[0], gbhh[0],
                                                   gWhh[1], gbhh[1], gx1);
    // ---- GRU layer 1 (input gx1, K=512)
    k_gemm<true, true, false, 0, 4><<<dim3(CBS / 16, 3), blk128, 0, stream>>>(
        gx1, gWih[2], gbih[2], nullptr, xpF, CH, CH, CH, 768);
    k_gemm<true, true, false, 0, 4><<<dim3(CBS / 16, 3), blk128, 0, stream>>>(
        gx1, gWih[3], gbih[3], nullptr, xpB, CH, CH, CH, 768);
    k_gru_scan<<<dim3(CB, 2), blk256, 0, stream>>>(xpF, xpB, gWhh[2], gbhh[2],
                                                   gWhh[3], gbhh[3], gx2);

    // ---- relative position machinery
    k_layernorm<<<512, blk256, 0, stream>>>(rel_emb, relln, rel_g, rel_b);
    k_relpos<<<(512 * 512) / 256, blk256, 0, stream>>>(relpos);

    // ---- transformer layers
    for (int L = 0; L < 2; ++L) {
        const float* P[16];
        for (int i = 0; i < 16; ++i) P[i] = (const float*)d_in[20 + 16 * L + i];
        const float *qw = P[0], *qb = P[1], *kw = P[2], *kb2 = P[3];
        const float *vw = P[4], *vb = P[5], *ow = P[6], *ob = P[7];
        const float *l1g = P[8], *l1b = P[9], *iw = P[10], *ib = P[11];
        const float *o2w = P[12], *o2b = P[13], *l2g = P[14], *l2b = P[15];

        const float* hin = (L == 0) ? gx2 : hbuf;

        // projections (N=512 -> grid.y=2)
        k_gemm<false, true, false, 0, 4><<<dim3(CBS / 16, 2), blk128, 0, stream>>>(
            hin, qw, qb, nullptr, q, CH, CH, CH, CH);
        k_gemm<false, true, false, 0, 4><<<dim3(CBS / 16, 2), blk128, 0, stream>>>(
            hin, kw, kb2, nullptr, kk, CH, CH, CH, CH);
        k_gemm<false, true, false, 0, 4><<<dim3(CBS / 16, 2), blk128, 0, stream>>>(
            hin, vw, vb, nullptr, v, CH, CH, CH, CH);
        k_gemm<false, true, false, 0, 4><<<dim3(512 / 16, 2), blk128, 0, stream>>>(
            relln, kw, kb2, nullptr, posk, CH, CH, CH, CH);
        k_gemm<false, true, false, 0, 4><<<dim3(512 / 16, 2), blk128, 0, stream>>>(
            relln, qw, qb, nullptr, posq, CH, CH, CH, CH);

        k_qpos<<<dim3(32, 2, CB * CNH), blk128, 0, stream>>>(q, posk, c2p);
        k_qpos<<<dim3(32, 2, CB * CNH), blk128, 0, stream>>>(kk, posq, p2c);
        k_scores<<<dim3(32, 2, CB * CNH), blk128, 0, stream>>>(q, kk, relpos, c2p, p2c, scores);
        k_softmax<<<CB * CNH * CS, blk512, 0, stream>>>(scores);
        k_ctx<<<dim3(32, 1, CB * CNH), blk128, 0, stream>>>(scores, v, ctx);

        // o-projection + residual, LN1
        k_gemm<false, true, true, 0, 4><<<dim3(CBS / 16, 2), blk128, 0, stream>>>(
            ctx, ow, ob, hin, y1, CH, CH, CH, CH);
        k_layernorm<<<CBS, blk256, 0, stream>>>(y1, a1, l1g, l1b);

        // FFN: gelu(a1@iw+ib) (N=256 -> grid.y=1), then @o2w + o2b + a1
        k_gemm<false, true, false, 1, 4><<<dim3(CBS / 16, 1), blk128, 0, stream>>>(
            a1, iw, ib, nullptr, ff, CH, CH, CINT, CINT);
        k_gemm<false, true, true, 0, 4><<<dim3(CBS / 16, 2), blk128, 0, stream>>>(
            ff, o2w, o2b, a1, y2, CINT, CINT, CH, CH);
        float* lnout = (L == 0) ? hbuf : (float*)d_out;
        k_layernorm<<<CBS, blk256, 0, stream>>>(y2, lnout, l2g, l2b);
    }
    (void)in_sizes; (void)n_in; (void)out_size; (void)ws_size;
}